// CosSim2D_57492432224258
// MI455X (gfx1250) — compile-verified
//
#include <hip/hip_runtime.h>
#include <math.h>

typedef __attribute__((ext_vector_type(2))) float v2f;
typedef __attribute__((ext_vector_type(8))) float v8f;

#define KSZ    3
#define BATCH  32
#define Hdim   128
#define Wdim   128
#define Cdim   64
#define Udim   256
#define OY     126
#define OX     126
#define Fpos   (OY*OX)          // 15876 positions per batch
#define Ddim   (KSZ*KSZ*Cdim)   // 576
#define TILE_M 128
#define TILE_N 64
#define KBLK   32
#define NKB    (Ddim/KBLK)      // 18
#define ASTRIDE (KBLK + 4)      // conflict-free b64 frag loads (36)
#define BSTRIDE (KBLK + 2)      // (34)

#if __has_builtin(__builtin_amdgcn_global_load_async_to_lds_b32) && \
    __has_builtin(__builtin_amdgcn_s_wait_asynccnt)
#define HAVE_ASYNC_LDS 1
typedef __attribute__((address_space(1))) int* gas_ptr;   // '__device__ int *'
typedef __attribute__((address_space(3))) int* las_ptr;   // '__shared__ int *'
#endif

// ---------------------------------------------------------------------------
// Prep: w_scaled[d][u] = w[d][u] / (sqrt(max(sum_d w^2,1e-12)) + (q/0.3)^2)
//       pe[u] = (p[u]/5)^2.  One block, 256 threads (thread == unit u).
// ---------------------------------------------------------------------------
__global__ __launch_bounds__(256)
void cossim_prep_kernel(const float* __restrict__ w, const float* __restrict__ p,
                        const float* __restrict__ q, float* __restrict__ w_s,
                        float* __restrict__ pe) {
  const int u = threadIdx.x;
  float qq = q[0] / 0.3f; qq = qq * qq;
  float ss = 0.f;
  for (int d = 0; d < Ddim; ++d) { float v = w[d * Udim + u]; ss = fmaf(v, v, ss); }
  const float inv = 1.f / (sqrtf(fmaxf(ss, 1e-12f)) + qq);
  for (int d = 0; d < Ddim; ++d) w_s[d * Udim + u] = w[d * Udim + u] * inv;
  const float pp = p[u] / 5.0f;
  pe[u] = pp * pp;
}

// ---------------------------------------------------------------------------
// Fused im2col + fp32-WMMA GEMM + signed-power epilogue.
// Block = 256 threads = 8 waves; block tile 128(M) x 64(N).
// Waves arranged 4(M) x 2(N); wave tile 32x32 = 4 accumulators, so each
// A/B fragment feeds 2 WMMAs (0.5 ds_load_b64 per v_wmma).
// A is staged through VGPRs (gives patch sum-of-squares for free);
// B is staged with GLOBAL_LOAD_ASYNC_TO_LDS_B32 per-lane scatter (transpose
// done in the DMA addressing) when the toolchain exposes the builtin.
// ---------------------------------------------------------------------------
__global__ __launch_bounds__(256)
void cossim_gemm_kernel(const float* __restrict__ x, const float* __restrict__ w_s,
                        const float* __restrict__ pe, const float* __restrict__ q,
                        float* __restrict__ out) {
  __shared__ float As[TILE_M][ASTRIDE];   // [m][k]
  __shared__ float Bs[TILE_N][BSTRIDE];   // transposed [n][k]
  __shared__ float xnsum[TILE_M];
  __shared__ float xnorm[TILE_M];

  const int tid   = threadIdx.x;
  const int lane  = tid & 31;
  const int wave  = tid >> 5;
  const int b     = blockIdx.z;
  const int nbase = blockIdx.y * TILE_N;
  const int mbase = blockIdx.x * TILE_M;

  // --- A staging map: 2 threads per patch row, 16 contiguous channels each ---
  const int arow = tid >> 1;            // 0..127
  const int acq  = (tid & 1) * 16;      // 0 or 16 within the 32-chan slice
  int f_ld = mbase + arow;
  if (f_ld >= Fpos) f_ld = Fpos - 1;    // clamp (stores guarded later)
  const int py = f_ld / OX, px = f_ld % OX;
  const float* xb = x + (size_t)b * Hdim * Wdim * Cdim;

  // --- B staging map: element e = op*256 + tid of the 32x64 slice ---
  const int bkr = tid >> 6;             // k sub-row 0..3 (advances by 4 per op)
  const int bnc = tid & 63;             // unit column 0..63

  if (tid < TILE_M) xnsum[tid] = 0.f;

  // --- wave tile coordinates ---
  const int m0 = (wave >> 1) * 32;
  const int n0 = (wave & 1) * 32;
  const int lm = lane & 15;
  const int lh = lane >> 4;             // lane-half: K pair select / M+8 select

  v8f acc00 = {}, acc01 = {}, acc10 = {}, acc11 = {};
  float xs = 0.f;                       // per-thread partial patch sumsq

  for (int kb = 0; kb < NKB; ++kb) {
    const int pos = kb >> 1, half = kb & 1;
    const int ii = pos / 3, jj = pos % 3;
    const float* asrc =
        xb + (((size_t)(py + ii) * Wdim) + (px + jj)) * Cdim + half * 32 + acq;
    const float4 a0 = *(const float4*)asrc;
    const float4 a1 = *(const float4*)(asrc + 4);
    const float4 a2 = *(const float4*)(asrc + 8);
    const float4 a3 = *(const float4*)(asrc + 12);

    __syncthreads();  // previous tile fully consumed by all waves

#if defined(HAVE_ASYNC_LDS)
    {
      // per-lane async scatter: w_s[k0+k][nbase+bnc] -> Bs[bnc][k], k = op*4+bkr
      const float* gsrc = w_s + (size_t)(kb * KBLK + bkr) * Udim + nbase + bnc;
      float*       ldst = &Bs[bnc][bkr];
#pragma unroll
      for (int op = 0; op < 8; ++op) {
        __builtin_amdgcn_global_load_async_to_lds_b32(
            (gas_ptr)(__UINTPTR_TYPE__)gsrc,
            (las_ptr)(unsigned)(__UINTPTR_TYPE__)ldst, 0, 0);
        gsrc += 4 * Udim;   // k advances by 4 rows per op
        ldst += 4;          // transposed LDS layout: +4 floats along k
      }
    }
#endif

    *(float4*)&As[arow][acq]      = a0;
    *(float4*)&As[arow][acq + 4]  = a1;
    *(float4*)&As[arow][acq + 8]  = a2;
    *(float4*)&As[arow][acq + 12] = a3;
    xs += a0.x*a0.x + a0.y*a0.y + a0.z*a0.z + a0.w*a0.w
        + a1.x*a1.x + a1.y*a1.y + a1.z*a1.z + a1.w*a1.w
        + a2.x*a2.x + a2.y*a2.y + a2.z*a2.z + a2.w*a2.w
        + a3.x*a3.x + a3.y*a3.y + a3.z*a3.z + a3.w*a3.w;

#if !defined(HAVE_ASYNC_LDS)
    {
      // synchronous fallback: gather 8 strided w_s values, transpose into LDS
#pragma unroll
      for (int op = 0; op < 2; ++op) {
        const float* bs = w_s + (size_t)(kb * KBLK + bkr + op * 4) * Udim + nbase + bnc;
        Bs[bnc][bkr + op * 4]      = bs[0];
        Bs[bnc][bkr + op * 4 + 8]  = bs[8 * Udim];
        Bs[bnc][bkr + op * 4 + 16] = bs[16 * Udim];
        Bs[bnc][bkr + op * 4 + 24] = bs[24 * Udim];
      }
    }
#endif

    // L2 prefetch of the next A slice (global_prefetch_b8)
    if (kb + 1 < NKB) {
      const int np = (kb + 1) >> 1, nh = (kb + 1) & 1;
      const float* pf =
          xb + (((size_t)(py + np / 3) * Wdim) + (px + np % 3)) * Cdim + nh * 32 + acq;
      __builtin_prefetch(pf, 0, 0);
    }

#if defined(HAVE_ASYNC_LDS)
    __builtin_amdgcn_s_wait_asynccnt(0);  // this wave's async LDS writes done
#endif
    __syncthreads();  // tile visible to every wave

#pragma unroll
    for (int kk = 0; kk < KBLK; kk += 4) {
      const int kp = kk + (lh << 1);  // lanes 0-15: K{kk,kk+1}; 16-31: K{kk+2,kk+3}
      v2f af0 = *(const v2f*)&As[m0 + lm][kp];
      v2f af1 = *(const v2f*)&As[m0 + 16 + lm][kp];
      v2f bf0 = *(const v2f*)&Bs[n0 + lm][kp];
      v2f bf1 = *(const v2f*)&Bs[n0 + 16 + lm][kp];
      acc00 = __builtin_amdgcn_wmma_f32_16x16x4_f32(false, af0, false, bf0,
                                                    (short)0, acc00, false, false);
      acc01 = __builtin_amdgcn_wmma_f32_16x16x4_f32(false, af0, false, bf1,
                                                    (short)0, acc01, false, false);
      acc10 = __builtin_amdgcn_wmma_f32_16x16x4_f32(false, af1, false, bf0,
                                                    (short)0, acc10, false, false);
      acc11 = __builtin_amdgcn_wmma_f32_16x16x4_f32(false, af1, false, bf1,
                                                    (short)0, acc11, false, false);
    }
  }

  // --- reduce patch sumsq (2 partials/row) and build 1/(||x||+q^2) ---
  atomicAdd(&xnsum[arow], xs);
  __syncthreads();
  if (tid < TILE_M) {
    float qq = q[0] / 0.3f; qq = qq * qq;
    xnorm[tid] = 1.f / (sqrtf(fmaxf(xnsum[tid], 1e-12f)) + qq);
  }
  __syncthreads();

  // --- epilogue: sign(s) * (|s|+eps)^pe[n] ---
  const int ng0 = nbase + n0 + lm;
  const int ng1 = ng0 + 16;
  const float pe0 = pe[ng0];
  const float pe1 = pe[ng1];
#pragma unroll
  for (int r = 0; r < 8; ++r) {
    const int mA = m0 + r + lh * 8;    // C/D layout: VGPR r -> M=r (lanes<16) / r+8
    const int mB = mA + 16;
    const int fA = mbase + mA;
    const int fB = mbase + mB;
    if (fA < Fpos) {
      const float invn = xnorm[mA];
      const float t0 = acc00[r] * invn;
      const float t1 = acc01[r] * invn;
      const float s0 = (t0 > 0.f) ? 1.f : ((t0 < 0.f) ? -1.f : 0.f);
      const float s1 = (t1 > 0.f) ? 1.f : ((t1 < 0.f) ? -1.f : 0.f);
      float* orow = out + ((size_t)b * Fpos + fA) * Udim;
      orow[ng0] = s0 * __powf(fabsf(t0) + 1e-6f, pe0);
      orow[ng1] = s1 * __powf(fabsf(t1) + 1e-6f, pe1);
    }
    if (fB < Fpos) {
      const float invn = xnorm[mB];
      const float t0 = acc10[r] * invn;
      const float t1 = acc11[r] * invn;
      const float s0 = (t0 > 0.f) ? 1.f : ((t0 < 0.f) ? -1.f : 0.f);
      const float s1 = (t1 > 0.f) ? 1.f : ((t1 < 0.f) ? -1.f : 0.f);
      float* orow = out + ((size_t)b * Fpos + fB) * Udim;
      orow[ng0] = s0 * __powf(fabsf(t0) + 1e-6f, pe0);
      orow[ng1] = s1 * __powf(fabsf(t1) + 1e-6f, pe1);
    }
  }
}

// ---------------------------------------------------------------------------
extern "C" void kernel_launch(void* const* d_in, const int* in_sizes, int n_in,
                              void* d_out, int out_size, void* d_ws, size_t ws_size,
                              hipStream_t stream) {
  (void)in_sizes; (void)n_in; (void)out_size; (void)ws_size;
  const float* x = (const float*)d_in[0];   // (32,128,128,64)
  const float* w = (const float*)d_in[1];   // (1,576,256)
  const float* p = (const float*)d_in[2];   // (256,)
  const float* q = (const float*)d_in[3];   // (1,)
  float* out = (float*)d_out;               // (32,126,126,256)

  float* wsf = (float*)d_ws;
  float* w_s = wsf;                         // 576*256 floats
  float* pe  = wsf + Ddim * Udim;           // 256 floats

  cossim_prep_kernel<<<1, 256, 0, stream>>>(w, p, q, w_s, pe);

  dim3 grid((Fpos + TILE_M - 1) / TILE_M,   // 125 M tiles
            Udim / TILE_N,                  // 4 N tiles
            BATCH);                         // 32
  cossim_gemm_kernel<<<grid, 256, 0, stream>>>(x, w_s, pe, q, out);
}